// AOLayer_18451179503906
// MI455X (gfx1250) — compile-verified
//
#include <hip/hip_runtime.h>

typedef float v2f __attribute__((ext_vector_type(2)));
typedef float v8f __attribute__((ext_vector_type(8)));

#define B_   512
#define N_   32
#define A_   256
#define P_   6
#define ROWS (B_ * N_)   // 16384 rows of the (ROWS x A_) output

__device__ __forceinline__ float ang_term(int pw, float d) {
    // pw==0 -> 1, pw==1 -> d, pw==2 -> d*d   (branchless -> v_cndmask)
    float dd = d * d;
    float t  = (pw == 1) ? d : dd;
    return (pw == 0) ? 1.0f : t;
}

__global__ __launch_bounds__(256) void ao_layer_wmma_kernel(
    const float* __restrict__ pos,      // (ROWS, 3)
    const float* __restrict__ centers,  // (A_, 3)
    const float* __restrict__ exps,     // (A_, P_)
    const float* __restrict__ coeffs,   // (A_, P_)
    const int*   __restrict__ powers,   // (A_, 3)
    float*       __restrict__ out)      // (ROWS, A_)
{
    const int lane = threadIdx.x & 31;
    const int wave = threadIdx.x >> 5;
    const int tile = blockIdx.x * 8 + wave;   // 16384 tiles total, exact
    const int rowBase = (tile >> 4) << 4;     // 1024 row tiles
    const int colBase = (tile & 15) << 4;     // 16 col tiles

    const int lm = lane & 15;
    const int hi = lane >> 4;                 // 0: K=0,1 half   1: K=2,3 half

    // ---------------- per-lane column (basis function) data ----------------
    const int col = colBase + lm;
    const float cx = centers[col * 3 + 0];
    const float cy = centers[col * 3 + 1];
    const float cz = centers[col * 3 + 2];
    const float csq = cx * cx + cy * cy + cz * cz;
    const int pwx = powers[col * 3 + 0];
    const int pwy = powers[col * 3 + 1];
    const int pwz = powers[col * 3 + 2];
    float e[P_], cf[P_];
#pragma unroll
    for (int p = 0; p < P_; ++p) {
        e[p]  = exps[col * P_ + p];
        cf[p] = coeffs[col * P_ + p];
    }

    // ---- per-lane row data: the 8 rows this lane's accumulators cover ----
    // C/D layout: VGPR v holds row m = v + 8*hi, column lm.
    float prx[8], pry[8], prz[8];
    v8f cacc;
#pragma unroll
    for (int v = 0; v < 8; ++v) {
        const int r = rowBase + v + 8 * hi;
        prx[v] = pos[r * 3 + 0];
        pry[v] = pos[r * 3 + 1];
        prz[v] = pos[r * 3 + 2];
        cacc[v] = prx[v] * prx[v] + pry[v] * pry[v] + prz[v] * prz[v]; // |p_m|^2
    }

    // ---- A fragment (16x4 f32): row m = lm; A[m, 0..3] = [x, y, z, 1] ----
    const int arow = rowBase + lm;
    const float ax = pos[arow * 3 + 0];
    const float ay = pos[arow * 3 + 1];
    const float az = pos[arow * 3 + 2];
    v2f afrag;
    afrag.x = hi ? az   : ax;   // K = 2*hi + 0
    afrag.y = hi ? 1.0f : ay;   // K = 2*hi + 1

    // ---- B fragment (4x16 f32): col n = lm; B[0..3, n] = [-2cx,-2cy,-2cz,|c|^2]
    v2f bfrag;
    bfrag.x = hi ? (-2.0f * cz) : (-2.0f * cx);
    bfrag.y = hi ? csq          : (-2.0f * cy);

    // r2[m][n] = |p_m|^2 - 2 p_m . c_n + |c_n|^2  via v_wmma_f32_16x16x4_f32
    v8f r2 = __builtin_amdgcn_wmma_f32_16x16x4_f32(
        /*neg_a=*/false, afrag, /*neg_b=*/false, bfrag,
        /*c_mod=*/(short)0, cacc, /*reuse_a=*/false, /*reuse_b=*/false);

    // ------------------------- epilogue per element -------------------------
#pragma unroll
    for (int v = 0; v < 8; ++v) {
        const float q = r2[v];
        float rad = 0.0f;
#pragma unroll
        for (int p = 0; p < P_; ++p)
            rad = __builtin_fmaf(cf[p], __expf(-e[p] * q), rad);

        const float dx = prx[v] - cx;
        const float dy = pry[v] - cy;
        const float dz = prz[v] - cz;
        const float ang = ang_term(pwx, dx) * ang_term(pwy, dy) * ang_term(pwz, dz);

        out[(size_t)(rowBase + v + 8 * hi) * A_ + col] = ang * rad;
    }
}

extern "C" void kernel_launch(void* const* d_in, const int* in_sizes, int n_in,
                              void* d_out, int out_size, void* d_ws, size_t ws_size,
                              hipStream_t stream) {
    const float* pos     = (const float*)d_in[0];
    const float* centers = (const float*)d_in[1];
    const float* exps    = (const float*)d_in[2];
    const float* coeffs  = (const float*)d_in[3];
    const int*   powers  = (const int*)d_in[4];
    float* out = (float*)d_out;

    const int tiles = (ROWS / 16) * (A_ / 16);  // 16384, 8 waves per block
    dim3 grid(tiles / 8), block(256);
    ao_layer_wmma_kernel<<<grid, block, 0, stream>>>(pos, centers, exps, coeffs,
                                                     powers, out);
}